// RelGraphConv_64493228917360
// MI455X (gfx1250) — compile-verified
//
#include <hip/hip_runtime.h>

typedef __attribute__((ext_vector_type(2))) float v2f;
typedef __attribute__((ext_vector_type(8))) float v8f;

#define NNODES 100000
#define NRELS  8
#define NEDGES 300000
#define DDIM   128
#define NBASES 4
#define NTILES (NNODES / 16)   // 6250, exact

#if defined(__HIP_PLATFORM_AMD__)
#define ATOMIC_ADD_F32 unsafeAtomicAdd
#else
#define ATOMIC_ADD_F32 atomicAdd
#endif

// ---------------------------------------------------------------------------
// W[r] = sum_b coeff[r,b] * basis[b]   ->  W: [NRELS, DDIM, DDIM] in ws
// ---------------------------------------------------------------------------
__global__ void compute_w_kernel(const float* __restrict__ coeff,
                                 const float* __restrict__ basis,
                                 float* __restrict__ W) {
    int i = blockIdx.x * blockDim.x + threadIdx.x;
    if (i >= NRELS * DDIM * DDIM) return;
    int r  = i / (DDIM * DDIM);
    int io = i % (DDIM * DDIM);
    float s = 0.f;
#pragma unroll
    for (int b = 0; b < NBASES; ++b)
        s += coeff[r * NBASES + b] * basis[b * DDIM * DDIM + io];
    W[i] = s;
}

// ---------------------------------------------------------------------------
// Edge phase for one relation: 32 lanes per edge.
//   agg[dst] += x[src]  (float4 per lane, 4 fp32 atomics)
//   deg[dst] += 1       (lane 0)
// agg fits in L2 (51.2MB < 192MB) so atomics resolve in-cache.
// ---------------------------------------------------------------------------
__global__ void edge_scatter_kernel(const float* __restrict__ x,
                                    const int* __restrict__ src,
                                    const int* __restrict__ dst,
                                    float* __restrict__ agg,
                                    float* __restrict__ deg) {
    long gid = (long)blockIdx.x * blockDim.x + threadIdx.x;
    int e = (int)(gid >> 5);
    int q = (int)(gid & 31);
    if (e >= NEDGES) return;
    int s = src[e];
    int d = dst[e];
    const float4* xr = (const float4*)(x + (size_t)s * DDIM);
    float4 xv = xr[q];
    float* a = agg + (size_t)d * DDIM + q * 4;
    ATOMIC_ADD_F32(a + 0, xv.x);
    ATOMIC_ADD_F32(a + 1, xv.y);
    ATOMIC_ADD_F32(a + 2, xv.z);
    ATOMIC_ADD_F32(a + 3, xv.w);
    if (q == 0) ATOMIC_ADD_F32(deg + d, 1.0f);
}

// ---------------------------------------------------------------------------
// out[M,O] (+)= (use_deg ? A/deg : A) @ Wg   via V_WMMA_F32_16X16X4_F32
// If do_relu: out = relu(out_acc + bias)
//
// 256 threads = 8 waves; each wave: one 16-row node tile x 128 cols (8 C tiles).
//
// W is staged into LDS in a swizzled layout so each B fragment is one
// conflict-free ds_load_b64:
//   element (k, n) -> Wlds[(k>>1)*256 + ((2n + ((k>>1)&1)*32) & 255) + (k&1)]
// Load for (pair = kb>>1, col): b64 at pair*256 + ((2*col + (pair&1)*32)&255).
// The 32-float rotation on odd pair-rows puts lanes 0-15 and 16-31 on
// disjoint 32-bank halves (pair parity == lane-half), so all 64 banks are
// used with no conflicts.
//
// VGPR layouts (ISA 7.12.2, 32-bit):
//   A 16x4 : lane%16 = M; vgpr j holds K = 2*(lane/16) + j
//   B 4x16 : lane%16 = N; vgpr j holds K = 2*(lane/16) + j
//   C 16x16: lane%16 = N; vgpr v holds M = v + 8*(lane/16)
// ---------------------------------------------------------------------------
__global__ __launch_bounds__(256)
void rgcn_wmma_gemm_kernel(const float* __restrict__ A,
                           const float* __restrict__ Wg,
                           const float* __restrict__ deg,
                           const float* __restrict__ bias,
                           float* __restrict__ out,
                           int use_deg, int do_relu) {
    __shared__ float Wlds[DDIM * DDIM];     // 64 KB, swizzled

    // Cooperative swizzled W load: coalesced b128 global reads, scalar LDS
    // stores into the swizzle (one-time cost per block).
    {
        const float4* Wv = (const float4*)Wg;
        for (int i4 = threadIdx.x; i4 < DDIM * DDIM / 4; i4 += 256) {
            float4 w = Wv[i4];
            int k = (i4 * 4) >> 7;          // row (same for all 4 elements)
            int n0 = (i4 * 4) & 127;
            int pr = k >> 1;
            int rot = (pr & 1) * 32;
            int base = pr * 256;
            int kb1 = k & 1;
            Wlds[base + ((2 * (n0 + 0) + rot) & 255) + kb1] = w.x;
            Wlds[base + ((2 * (n0 + 1) + rot) & 255) + kb1] = w.y;
            Wlds[base + ((2 * (n0 + 2) + rot) & 255) + kb1] = w.z;
            Wlds[base + ((2 * (n0 + 3) + rot) & 255) + kb1] = w.w;
        }
    }
    __syncthreads();

    int wave = threadIdx.x >> 5;
    int lane = threadIdx.x & 31;
    int half = lane >> 4;        // 0 or 1
    int l16  = lane & 15;

    int tile = blockIdx.x * 8 + wave;       // 16-row node tile (wave-uniform)
    if (tile >= NTILES) return;             // EXEC stays all-ones per wave
    int rowBase = tile * 16;

    // Degree normalization folded into A fragment (A-row for this lane = l16)
    float rdeg = 1.0f;
    if (use_deg) rdeg = 1.0f / fmaxf(deg[rowBase + l16], 1.0f);

    // Per-lane B column offsets (loop-invariant): this lane's pair parity
    // always equals `half`, so the rotation is a lane constant.
    int colBase = 2 * l16 + 32 * half;      // rotation folded in
    int bOff[8];
#pragma unroll
    for (int nt = 0; nt < 8; ++nt)
        bOff[nt] = (colBase + 32 * nt) & 255;

    // Load C (accumulated output so far)
    v8f c[8];
#pragma unroll
    for (int nt = 0; nt < 8; ++nt) {
#pragma unroll
        for (int v = 0; v < 8; ++v)
            c[nt][v] = out[(size_t)(rowBase + v + 8 * half) * DDIM + nt * 16 + l16];
    }

    const float* Arow = A + (size_t)(rowBase + l16) * DDIM + 2 * half;

    // K loop: 4 chunks x 8 steps; batch the 8 A loads per chunk so the
    // scheduler can overlap them with WMMA issue (graduated s_wait_loadcnt).
#pragma unroll
    for (int kc = 0; kc < 4; ++kc) {
        v2f a[8];
#pragma unroll
        for (int j = 0; j < 8; ++j)
            a[j] = *(const v2f*)(Arow + (kc * 8 + j) * 4);   // 8B-aligned
#pragma unroll
        for (int j = 0; j < 8; ++j) {
            int ks = kc * 8 + j;
            int pair = ks * 2 + half;       // B pair-row for this lane
            const float* bRow = Wlds + pair * 256;
            v2f as = a[j] * rdeg;
#pragma unroll
            for (int nt = 0; nt < 8; ++nt) {
                v2f b = *(const v2f*)(bRow + bOff[nt]);      // ds_load_b64
                // (neg_a, A, neg_b, B, c_mod, C, reuse_a, reuse_b)
                c[nt] = __builtin_amdgcn_wmma_f32_16x16x4_f32(
                    false, as, false, b, (short)0, c[nt], false, false);
            }
        }
    }

#pragma unroll
    for (int nt = 0; nt < 8; ++nt) {
#pragma unroll
        for (int v = 0; v < 8; ++v) {
            float val = c[nt][v];
            if (do_relu) {
                val += bias[nt * 16 + l16];
                val = fmaxf(val, 0.0f);
            }
            out[(size_t)(rowBase + v + 8 * half) * DDIM + nt * 16 + l16] = val;
        }
    }
}

// ---------------------------------------------------------------------------
extern "C" void kernel_launch(void* const* d_in, const int* in_sizes, int n_in,
                              void* d_out, int out_size, void* d_ws, size_t ws_size,
                              hipStream_t stream) {
    (void)in_sizes; (void)n_in; (void)out_size; (void)ws_size;

    const float* x      = (const float*)d_in[0];   // [NNODES, DDIM]
    const int*   src    = (const int*)  d_in[1];   // [NRELS, NEDGES]
    const int*   dst    = (const int*)  d_in[2];   // [NRELS, NEDGES]
    const float* coeff  = (const float*)d_in[3];   // [NRELS, NBASES]
    const float* basis  = (const float*)d_in[4];   // [NBASES, DDIM, DDIM]
    const float* loop_w = (const float*)d_in[5];   // [DDIM, DDIM]
    const float* bias   = (const float*)d_in[6];   // [DDIM]
    float* out = (float*)d_out;                    // [NNODES, DDIM]

    // Workspace layout (~52.1 MB):
    float* W   = (float*)d_ws;                     // NRELS*DDIM*DDIM
    float* agg = W + (size_t)NRELS * DDIM * DDIM;  // NNODES*DDIM
    float* deg = agg + (size_t)NNODES * DDIM;      // NNODES  (contiguous after agg)

    // out accumulates across relation GEMMs -> start at zero
    hipMemsetAsync(out, 0, (size_t)NNODES * DDIM * sizeof(float), stream);

    // Basis-combined relation weights
    compute_w_kernel<<<(NRELS * DDIM * DDIM + 255) / 256, 256, 0, stream>>>(
        coeff, basis, W);

    const int gemm_blocks = (NTILES + 7) / 8;      // 782
    const long edge_threads = (long)NEDGES * 32;
    const int edge_blocks = (int)((edge_threads + 255) / 256);

    for (int r = 0; r < NRELS; ++r) {
        // zero agg + deg in one shot (contiguous)
        hipMemsetAsync(agg, 0,
                       ((size_t)NNODES * DDIM + NNODES) * sizeof(float), stream);
        edge_scatter_kernel<<<edge_blocks, 256, 0, stream>>>(
            x, src + (size_t)r * NEDGES, dst + (size_t)r * NEDGES, agg, deg);
        rgcn_wmma_gemm_kernel<<<gemm_blocks, 256, 0, stream>>>(
            agg, W + (size_t)r * DDIM * DDIM, deg, nullptr, out,
            /*use_deg=*/1, /*do_relu=*/0);
    }

    // Self-loop GEMM + bias + ReLU
    rgcn_wmma_gemm_kernel<<<gemm_blocks, 256, 0, stream>>>(
        x, loop_w, nullptr, bias, out, /*use_deg=*/0, /*do_relu=*/1);
}